// dctmodule2D_77386720739897
// MI455X (gfx1250) — compile-verified
//
#include <hip/hip_runtime.h>

// 2D DCT-II (ortho) of (16,3,1024,1024) fp32  ==  Y = D @ X @ D^T per image.
// Two batched 1024^3 fp32 GEMMs using V_WMMA_F32_16X16X4_F32, with K-pair
// interleaved LDS tiles (single ds_load_b64 per WMMA fragment) and a
// register-staged double-buffered pipeline.

#define DCT_N 1024
#define NN (DCT_N * DCT_N)
#define BM 128
#define BN 128
#define BK 16
#define NTILES (DCT_N / BK)
#define PSTRA 132  // float2 stride per k-pair row of A tile (264 fl ≡ 8 banks)
#define PSTRB 144  // float2 stride per k-pair row of B tile (288 fl ≡ 32 banks)

typedef float v2f __attribute__((ext_vector_type(2)));
typedef float v8f __attribute__((ext_vector_type(8)));

// ---------------------------------------------------------------------------
// Build D[k][n] = c_k * cos(pi*(2n+1)*k / (2N)) and DT[n][k] = D[k][n].
// Integer phase reduction mod 4N keeps the cos argument in [0, 2*pi).
// ---------------------------------------------------------------------------
__global__ void build_dct_mats(float* __restrict__ D, float* __restrict__ DT) {
    int idx = blockIdx.x * blockDim.x + threadIdx.x;  // 0 .. N*N-1
    int k = idx >> 10;
    int n = idx & (DCT_N - 1);
    int m = (k * (2 * n + 1)) & (4 * DCT_N - 1);      // mod 4N -> arg < 2*pi
    float phase = (float)m * (3.14159265358979323846f / (2.0f * (float)DCT_N));
    float ck = (k == 0) ? 0.03125f                    // 1/sqrt(1024)
                        : 0.04419417382415922f;       // sqrt(2)/32
    float v = ck * cosf(phase);
    D[k * DCT_N + n]  = v;
    DT[n * DCT_N + k] = v;
}

// ---------------------------------------------------------------------------
// C[z] = A[z] @ B[z] for 1024x1024x1024 fp32; strides select batched/shared
// operands. 256 threads = 8 waves; block tile 128x128; wave tile 32x64.
// LDS tiles are K-pair interleaved: tile2[p][x] = {T[2p][x], T[2p+1][x]}, so
// each WMMA fragment (K = 2*half + {0,1}) is one aligned 8-byte LDS load.
// ---------------------------------------------------------------------------
__global__ __launch_bounds__(256)
void dct_gemm_f32(const float* __restrict__ Abase, long long strideA,
                  const float* __restrict__ Bbase, long long strideB,
                  float* __restrict__ Cbase, long long strideC) {
    __shared__ v2f As2[2][(BK / 2) * PSTRA];
    __shared__ v2f Bs2[2][(BK / 2) * PSTRB];

    const float* A = Abase + (long long)blockIdx.z * strideA;
    const float* B = Bbase + (long long)blockIdx.z * strideB;
    float*       C = Cbase + (long long)blockIdx.z * strideC;

    const int tid  = threadIdx.x;
    const int lane = tid & 31;
    const int wave = tid >> 5;          // 0..7
    const int waveM = wave & 3;         // 4 waves along M
    const int waveN = wave >> 2;        // 2 waves along N
    const int m0 = blockIdx.y * BM;
    const int n0 = blockIdx.x * BN;
    const int wm = waveM * 32;          // wave tile: 32 rows
    const int wn = waveN * 64;          // wave tile: 64 cols
    const int half = lane >> 4;         // 0: lanes 0-15, 1: lanes 16-31
    const int l16  = lane & 15;

    // global->LDS staging indices
    const int lk  = tid & 15;           // k within A tile (coalesced)
    const int lm  = tid >> 4;           // m base (stepped by 16)
    const int bn  = tid & 127;          // n within B tile (coalesced)
    const int bkk = tid >> 7;           // 0..1 (k, stepped by 2)

    v8f acc[2][4];
#pragma unroll
    for (int i = 0; i < 2; ++i)
#pragma unroll
        for (int j = 0; j < 4; ++j) acc[i][j] = v8f{};

    float areg[8], breg[8];

    auto load_global = [&](int k0) {
#pragma unroll
        for (int r = 0; r < 8; ++r)
            areg[r] = A[(long long)(m0 + lm + 16 * r) * DCT_N + (k0 + lk)];
#pragma unroll
        for (int r = 0; r < 8; ++r)
            breg[r] = B[(long long)(k0 + bkk + 2 * r) * DCT_N + (n0 + bn)];
    };
    auto store_lds = [&](int buf) {
        float* asf = (float*)&As2[buf][0];
        float* bsf = (float*)&Bs2[buf][0];
#pragma unroll
        for (int r = 0; r < 8; ++r)   // pair p = lk>>1, parity = lk&1
            asf[(lk >> 1) * (2 * PSTRA) + (lm + 16 * r) * 2 + (lk & 1)] = areg[r];
#pragma unroll
        for (int r = 0; r < 8; ++r)   // pair p = r, parity = bkk
            bsf[r * (2 * PSTRB) + bn * 2 + bkk] = breg[r];
    };

    // prologue: tile 0
    load_global(0);
    store_lds(0);
    __syncthreads();

    for (int t = 0; t < NTILES; ++t) {
        const int buf = t & 1;
        if (t + 1 < NTILES) load_global((t + 1) * BK);  // prefetch into VGPRs

        const v2f* ap = &As2[buf][0];
        const v2f* bp = &Bs2[buf][0];
#pragma unroll
        for (int kk = 0; kk < BK; kk += 4) {
            const int p = (kk >> 1) + half;             // K = 2p + {0,1}
            v2f afrag[2], bfrag[4];
#pragma unroll
            for (int i = 0; i < 2; ++i)
                afrag[i] = ap[p * PSTRA + (wm + 16 * i + l16)];
#pragma unroll
            for (int j = 0; j < 4; ++j)
                bfrag[j] = bp[p * PSTRB + (wn + 16 * j + l16)];
#pragma unroll
            for (int i = 0; i < 2; ++i)
#pragma unroll
                for (int j = 0; j < 4; ++j)
                    acc[i][j] = __builtin_amdgcn_wmma_f32_16x16x4_f32(
                        false, afrag[i], false, bfrag[j],
                        (short)0, acc[i][j], false, false);
        }

        if (t + 1 < NTILES) store_lds((t + 1) & 1);     // fill other buffer
        __syncthreads();
    }

    // store: vgpr r -> row (r + 8*half), col l16 within each 16x16 subtile
#pragma unroll
    for (int i = 0; i < 2; ++i) {
#pragma unroll
        for (int j = 0; j < 4; ++j) {
            const int crow = m0 + wm + i * 16 + half * 8;
            const int ccol = n0 + wn + j * 16 + l16;
#pragma unroll
            for (int r = 0; r < 8; ++r)
                C[(long long)(crow + r) * DCT_N + ccol] = acc[i][j][r];
        }
    }
}

// ---------------------------------------------------------------------------
extern "C" void kernel_launch(void* const* d_in, const int* in_sizes, int n_in,
                              void* d_out, int out_size, void* d_ws, size_t ws_size,
                              hipStream_t stream) {
    const float* x = (const float*)d_in[0];
    float* y = (float*)d_out;

    const int batch = 16 * 3;                 // 48 images of 1024x1024
    const size_t matBytes = (size_t)NN * sizeof(float);

    float* D  = (float*)d_ws;                 // [0, 4MB)
    float* DT = D + NN;                       // [4MB, 8MB)
    float* T  = DT + NN;                      // intermediate, chunked

    size_t avail = (ws_size > 2 * matBytes) ? (ws_size - 2 * matBytes) : 0;
    int chunk = (int)(avail / matBytes);
    if (chunk < 1) chunk = 1;
    if (chunk > batch) chunk = batch;

    build_dct_mats<<<NN / 256, 256, 0, stream>>>(D, DT);

    for (int b0 = 0; b0 < batch; b0 += chunk) {
        const int nb = (b0 + chunk <= batch) ? chunk : (batch - b0);
        dim3 grid(DCT_N / BN, DCT_N / BM, nb);  // (8, 8, nb)
        // Pass 1: T = X @ D^T   (B[k][n] = DT[k*N+n] = D[n][k])
        dct_gemm_f32<<<grid, 256, 0, stream>>>(
            x + (size_t)b0 * NN, (long long)NN,   // A batched
            DT, 0LL,                              // B shared
            T, (long long)NN);                    // C batched (chunk-local)
        // Pass 2: Y = D @ T
        dct_gemm_f32<<<grid, 256, 0, stream>>>(
            D, 0LL,                               // A shared
            T, (long long)NN,                     // B batched
            y + (size_t)b0 * NN, (long long)NN);  // C batched
    }
}